// GCNNet_4312147165192
// MI455X (gfx1250) — compile-verified
//
#include <hip/hip_runtime.h>
#include <math.h>

typedef __attribute__((ext_vector_type(2))) float v2f;
typedef __attribute__((ext_vector_type(8))) float v8f;

// ---------------------------------------------------------------------------
// Degree / normalization kernels
// ---------------------------------------------------------------------------
__global__ void k_deg_init(float* __restrict__ deg, int n) {
    int i = blockIdx.x * blockDim.x + threadIdx.x;
    if (i < n) deg[i] = 1.0f;  // self-loop contribution
}

__global__ void k_deg_count(const int* __restrict__ dst, float* __restrict__ deg, int E) {
    int e = blockIdx.x * blockDim.x + threadIdx.x;
    if (e < E) atomicAdd(&deg[dst[e]], 1.0f);
}

__global__ void k_deg_rsqrt(float* __restrict__ deg, int n) {
    int i = blockIdx.x * blockDim.x + threadIdx.x;
    if (i < n) {
        float d = deg[i];
        deg[i] = (d > 0.0f) ? rsqrtf(d) : 0.0f;
    }
}

// ---------------------------------------------------------------------------
// GEMM1: h1[N,16] = x[N,512] @ W1[512,16]   (fp32 WMMA 16x16x4)
// One wave computes one 16x16 output tile; K loop of 4.
// A 16x4 f32 layout: lanes 0-15 hold M=lane, VGPR0=K0,VGPR1=K1; lanes 16-31 K2,K3.
// B 4x16 f32 layout: symmetric (N in lane&15, K split across half-waves/VGPRs).
// C/D 16x16 f32: VGPR r -> M=r (lanes 0-15), M=8+r (lanes 16-31), N=lane&15.
// ---------------------------------------------------------------------------
__global__ void k_gemm1_wmma(const float* __restrict__ x, const float* __restrict__ W1,
                             float* __restrict__ h1, int n) {
    int wave   = (blockIdx.x * blockDim.x + threadIdx.x) >> 5;
    int lane   = threadIdx.x & 31;
    int ntiles = n >> 4;
    if (wave >= ntiles) return;  // wave-uniform: EXEC stays all-ones for WMMA

    int rowBase = wave << 4;
    int lm      = lane & 15;            // A row / B col within tile
    int kOff    = (lane >> 4) << 1;     // 0 or 2

    const float* xrow = x + (size_t)(rowBase + lm) * 512;

    v8f c = {0.f, 0.f, 0.f, 0.f, 0.f, 0.f, 0.f, 0.f};
    for (int k = 0; k < 512; k += 4) {
        v2f a = *(const v2f*)(xrow + k + kOff);  // 8B-aligned: k%4==0, kOff in {0,2}
        v2f b;
        b.x = W1[(size_t)(k + kOff)     * 16 + lm];
        b.y = W1[(size_t)(k + kOff + 1) * 16 + lm];
        c = __builtin_amdgcn_wmma_f32_16x16x4_f32(false, a, false, b,
                                                  (short)0, c, false, false);
    }

    int rowHalf = rowBase + ((lane >> 4) << 3);  // +0 or +8
#pragma unroll
    for (int r = 0; r < 8; ++r)
        h1[(size_t)(rowHalf + r) * 16 + lm] = c[r];
}

// ---------------------------------------------------------------------------
// Layer-1 aggregation
// ---------------------------------------------------------------------------
__global__ void k_agg1_init(const float* __restrict__ h1, const float* __restrict__ dinv,
                            const float* __restrict__ b1, float* __restrict__ agg, int n) {
    int t = blockIdx.x * blockDim.x + threadIdx.x;
    int i = t >> 4, f = t & 15;
    if (i < n) {
        float di = dinv[i];
        agg[(size_t)i * 16 + f] = h1[(size_t)i * 16 + f] * di * di + b1[f];
    }
}

__global__ void k_agg1_edges(const int* __restrict__ src, const int* __restrict__ dst,
                             const float* __restrict__ h1, const float* __restrict__ dinv,
                             float* __restrict__ agg, int E) {
    unsigned t = blockIdx.x * blockDim.x + threadIdx.x;
    unsigned e = t >> 4, f = t & 15;
    if (e < (unsigned)E) {
        int s = src[e], d = dst[e];
        float norm = dinv[s] * dinv[d];
        atomicAdd(&agg[(size_t)d * 16 + f], h1[(size_t)s * 16 + f] * norm);
    }
}

// ---------------------------------------------------------------------------
// W2 zero-pad to [16,48]
// ---------------------------------------------------------------------------
__global__ void k_w2_pad(const float* __restrict__ W2, float* __restrict__ W2p) {
    int t = blockIdx.x * blockDim.x + threadIdx.x;
    if (t < 16 * 48) {
        int i = t / 48, j = t % 48;
        W2p[t] = (j < 40) ? W2[i * 40 + j] : 0.0f;
    }
}

// ---------------------------------------------------------------------------
// GEMM2: h2[N,40] = relu(agg1)[N,16] @ W2p[16,48]  (3 col-tiles, K loop of 4)
// ---------------------------------------------------------------------------
__global__ void k_gemm2_wmma(const float* __restrict__ agg1, const float* __restrict__ W2p,
                             float* __restrict__ h2, int n) {
    int wave   = (blockIdx.x * blockDim.x + threadIdx.x) >> 5;
    int lane   = threadIdx.x & 31;
    int ntiles = n >> 4;
    if (wave >= ntiles) return;

    int rowBase = wave << 4;
    int lm      = lane & 15;
    int kOff    = (lane >> 4) << 1;

    const float* arow = agg1 + (size_t)(rowBase + lm) * 16;

    v8f acc[3];
#pragma unroll
    for (int t = 0; t < 3; ++t) acc[t] = (v8f){0.f,0.f,0.f,0.f,0.f,0.f,0.f,0.f};

#pragma unroll
    for (int k = 0; k < 16; k += 4) {
        v2f a;
        a.x = fmaxf(arow[k + kOff],     0.0f);  // fused ReLU
        a.y = fmaxf(arow[k + kOff + 1], 0.0f);
#pragma unroll
        for (int t = 0; t < 3; ++t) {
            v2f b;
            b.x = W2p[(size_t)(k + kOff)     * 48 + t * 16 + lm];
            b.y = W2p[(size_t)(k + kOff + 1) * 48 + t * 16 + lm];
            acc[t] = __builtin_amdgcn_wmma_f32_16x16x4_f32(false, a, false, b,
                                                           (short)0, acc[t], false, false);
        }
    }

    int rowHalf = rowBase + ((lane >> 4) << 3);
#pragma unroll
    for (int t = 0; t < 3; ++t) {
        int col = t * 16 + lm;
        if (col < 40) {
#pragma unroll
            for (int r = 0; r < 8; ++r)
                h2[(size_t)(rowHalf + r) * 40 + col] = acc[t][r];
        }
    }
}

// ---------------------------------------------------------------------------
// Layer-2 aggregation (into d_out) + log-softmax
// ---------------------------------------------------------------------------
__global__ void k_out_init(const float* __restrict__ h2, const float* __restrict__ dinv,
                           const float* __restrict__ b2, float* __restrict__ out, int n) {
    unsigned t = blockIdx.x * blockDim.x + threadIdx.x;
    unsigned i = t / 40u, f = t - i * 40u;
    if (i < (unsigned)n) {
        float di = dinv[i];
        out[(size_t)i * 40 + f] = h2[(size_t)i * 40 + f] * di * di + b2[f];
    }
}

__global__ void k_out_edges(const int* __restrict__ src, const int* __restrict__ dst,
                            const float* __restrict__ h2, const float* __restrict__ dinv,
                            float* __restrict__ out, int E) {
    unsigned t = blockIdx.x * blockDim.x + threadIdx.x;
    unsigned e = t / 40u, f = t - e * 40u;
    if (e < (unsigned)E) {
        int s = src[e], d = dst[e];
        float norm = dinv[s] * dinv[d];
        atomicAdd(&out[(size_t)d * 40 + f], h2[(size_t)s * 40 + f] * norm);
    }
}

__global__ void k_log_softmax40(float* __restrict__ out, int n) {
    int i = blockIdx.x * blockDim.x + threadIdx.x;
    if (i >= n) return;
    float* row = out + (size_t)i * 40;
    float v[40];
    float m = -INFINITY;
#pragma unroll
    for (int j = 0; j < 40; ++j) { v[j] = row[j]; m = fmaxf(m, v[j]); }
    float s = 0.0f;
#pragma unroll
    for (int j = 0; j < 40; ++j) s += expf(v[j] - m);
    float l = m + logf(s);
#pragma unroll
    for (int j = 0; j < 40; ++j) row[j] = v[j] - l;
}

// ---------------------------------------------------------------------------
// Host launcher
// ---------------------------------------------------------------------------
extern "C" void kernel_launch(void* const* d_in, const int* in_sizes, int n_in,
                              void* d_out, int out_size, void* d_ws, size_t ws_size,
                              hipStream_t stream) {
    const float* x  = (const float*)d_in[0];
    const int*   ei = (const int*)d_in[1];
    const float* W1 = (const float*)d_in[2];
    const float* b1 = (const float*)d_in[3];
    const float* W2 = (const float*)d_in[4];
    const float* b2 = (const float*)d_in[5];

    const int n = in_sizes[0] / 512;
    const int E = in_sizes[1] / 2;
    const int* src = ei;
    const int* dst = ei + E;

    // Workspace carve-out (256B aligned)
    char*  ws  = (char*)d_ws;
    size_t off = 0;
    auto alloc = [&](size_t bytes) -> void* {
        void* p = ws + off;
        off += (bytes + 255) & ~(size_t)255;
        return p;
    };
    float* dinv = (float*)alloc((size_t)n * 4);
    float* h1   = (float*)alloc((size_t)n * 16 * 4);
    float* agg1 = (float*)alloc((size_t)n * 16 * 4);
    float* h2   = (float*)alloc((size_t)n * 40 * 4);
    float* W2p  = (float*)alloc(16 * 48 * 4);
    float* out  = (float*)d_out;

    const int B = 256;
    auto blocks = [](long long work, int b) { return (unsigned)((work + b - 1) / b); };

    // 1) degrees -> dinv
    k_deg_init <<<blocks(n, B), B, 0, stream>>>(dinv, n);
    k_deg_count<<<blocks(E, B), B, 0, stream>>>(dst, dinv, E);
    k_deg_rsqrt<<<blocks(n, B), B, 0, stream>>>(dinv, n);

    // 2) GEMM1 (WMMA): h1 = x @ W1
    {
        long long threads = (long long)(n / 16) * 32;  // n divisible by 16 (100000)
        k_gemm1_wmma<<<blocks(threads, B), B, 0, stream>>>(x, W1, h1, n);
    }

    // 3) layer-1 aggregation
    k_agg1_init <<<blocks((long long)n * 16, B), B, 0, stream>>>(h1, dinv, b1, agg1, n);
    k_agg1_edges<<<blocks((long long)E * 16, B), B, 0, stream>>>(src, dst, h1, dinv, agg1, E);

    // 4) GEMM2 (WMMA, fused ReLU): h2 = relu(agg1) @ W2
    k_w2_pad<<<blocks(16 * 48, B), B, 0, stream>>>(W2, W2p);
    {
        long long threads = (long long)(n / 16) * 32;
        k_gemm2_wmma<<<blocks(threads, B), B, 0, stream>>>(agg1, W2p, h2, n);
    }

    // 5) layer-2 aggregation into d_out, then in-place log-softmax
    k_out_init <<<blocks((long long)n * 40, B), B, 0, stream>>>(h2, dinv, b2, out, n);
    k_out_edges<<<blocks((long long)E * 40, B), B, 0, stream>>>(src, dst, h2, dinv, out, E);
    k_log_softmax40<<<blocks(n, B), B, 0, stream>>>(out, n);
}